// GMMSegHead_2095944040758
// MI455X (gfx1250) — compile-verified
//
#include <hip/hip_runtime.h>
#include <math.h>

typedef float v2f __attribute__((ext_vector_type(2)));
typedef float v8f __attribute__((ext_vector_type(8)));

#define D       64
#define KCLS    19
#define PCOMP   5
#define NCOMP   95
#define NCOMP_P 96
#define TILE_N  128
#define THREADS 256
#define XT_STRIDE 66      // padded row stride for x-transpose tile (conflict-free)
#define LOGP_STRIDE 100   // padded per-pixel stride for logp spill (conflict-free)
#define W_FLOATS  12288   // 32 ksteps x 6 tiles x 32 lanes x 2

// shared layout (in floats):
//   [0,     8448)  xt   (128 rows x 66)           } reused after K-loop as logp
//   [8448, 20736)  W_pack (12288)                 }   spill: 8 x 16 x 100 = 12800
//   [20736,20832)  bias (96)
//   [20832,20896)  feat_w (64)
//   [20896,20960)  feat_b (64)
//   [20960,21728)  exchange scratch (3 x 256) for split LayerNorm
#define W_OFF     8448
#define BIAS_OFF  20736
#define FW_OFF    20832
#define FB_OFF    20896
#define EXA_OFF   20960
#define EXB_OFF   21216
#define EXC_OFF   21472
#define SH_FLOATS 21728

// ---------------------------------------------------------------------------
// One-time kernel: build packed GEMM weights + bias into workspace.
// logp_j(x) = sum_d x_d^2 * (-0.5*inv2) + sum_d x_d * (mu_hat*inv2) + bias_j
// Packed layout matches the WMMA B-fragment: lane l of kstep-block (kk, jtile)
// holds W[k = 4kk + 2*(l/16)+v][comp = 16*jtile + l%16] at [(kk*6+jt)*64 + 2l + v].
// ---------------------------------------------------------------------------
__global__ void build_w_kernel(const float* __restrict__ means,
                               const float* __restrict__ diag,
                               float* __restrict__ wout)   // [12288 W + 96 bias]
{
    const int j = threadIdx.x;
    if (j >= NCOMP_P) return;
    const int jt = j >> 4, li = j & 15;
    if (j < NCOMP) {
        const float* mu = means + j * D;
        const float* sc = diag  + j * D;
        float ss = 0.f;
        for (int d = 0; d < D; ++d) { float m = mu[d]; ss += m * m; }
        float rinv = 1.0f / fmaxf(sqrtf(ss), 1e-12f);       // l2_normalize(means)
        float qc = 0.f, logs = 0.f;
        for (int d = 0; d < D; ++d) {
            float s    = sc[d];
            float inv2 = 1.0f / (s * s);
            float mun  = mu[d] * rinv;
            qc   += mun * mun * inv2;
            logs += logf(s);
            int k = d;                                       // x^2 term
            int l = ((k & 3) >> 1) * 16 + li;
            wout[((k >> 2) * 6 + jt) * 64 + 2 * l + (k & 1)] = -0.5f * inv2;
            k = d + D;                                       // x term
            l = ((k & 3) >> 1) * 16 + li;
            wout[((k >> 2) * 6 + jt) * 64 + 2 * l + (k & 1)] = mun * inv2;
        }
        wout[W_FLOATS + j] = -0.5f * qc - logs - 58.81206412534085f; // 0.5*64*log(2pi)
    } else {                                                 // padding comp 95
        for (int k = 0; k < 2 * D; ++k) {
            int l = ((k & 3) >> 1) * 16 + li;
            wout[((k >> 2) * 6 + jt) * 64 + 2 * l + (k & 1)] = 0.f;
        }
        wout[W_FLOATS + j] = 0.f;
    }
}

__global__ __launch_bounds__(THREADS)
void gmm_seg_head_kernel(const float* __restrict__ base,   // [B, D, Npix]
                         const float* __restrict__ means,  // [K, P, D]
                         const float* __restrict__ diag,   // [K, P, D]
                         const float* __restrict__ feat_w, // [D]
                         const float* __restrict__ feat_b, // [D]
                         const float* __restrict__ mask_w, // [K]
                         const float* __restrict__ mask_b, // [K]
                         const float* __restrict__ wpre,   // precomputed W+bias (or null)
                         float* __restrict__ out,          // [B, K, Npix]
                         int Npix, int useWs)
{
    __shared__ float sh[SH_FLOATS];
    const int t    = threadIdx.x;
    const int lane = t & 31;
    const int wv   = t >> 5;
    const int tilesPerB = Npix / TILE_N;
    const int b  = blockIdx.x / tilesPerB;
    const int n0 = (blockIdx.x % tilesPerB) * TILE_N;

    // ---------- Phase 1: coalesced load of base[b][:, n0:n0+128], transposed into LDS
    {
        const float* src = base + (size_t)b * D * Npix + n0;
        #pragma unroll
        for (int i = 0; i < (D * TILE_N) / THREADS; ++i) {
            int idx = t + i * THREADS;
            int d = idx >> 7;            // / TILE_N
            int n = idx & (TILE_N - 1);
            sh[n * XT_STRIDE + d] = src[(size_t)d * Npix + n];
        }
        if (t < D) { sh[FW_OFF + t] = feat_w[t]; sh[FB_OFF + t] = feat_b[t]; }
    }

    // ---------- Phase 1b: bring in GMM weight matrix + bias
    if (useWs) {
        // fast path: copy precomputed packed W (L2-resident) into LDS
        const float4* wsrc = (const float4*)wpre;
        float4*       wdst = (float4*)(sh + W_OFF);
        #pragma unroll
        for (int i = 0; i < W_FLOATS / 4 / THREADS; ++i)
            wdst[t + i * THREADS] = wsrc[t + i * THREADS];
        if (t < NCOMP_P) sh[BIAS_OFF + t] = wpre[W_FLOATS + t];
    } else if (t < NCOMP_P) {
        // fallback: build in-block (identical math to build_w_kernel)
        const int j = t, jt = j >> 4, li = j & 15;
        if (j < NCOMP) {
            const float* mu = means + j * D;
            const float* sc = diag  + j * D;
            float ss = 0.f;
            for (int d = 0; d < D; ++d) { float m = mu[d]; ss += m * m; }
            float rinv = 1.0f / fmaxf(sqrtf(ss), 1e-12f);
            float qc = 0.f, logs = 0.f;
            for (int d = 0; d < D; ++d) {
                float s    = sc[d];
                float inv2 = 1.0f / (s * s);
                float mun  = mu[d] * rinv;
                qc   += mun * mun * inv2;
                logs += logf(s);
                int k = d;
                int l = ((k & 3) >> 1) * 16 + li;
                sh[W_OFF + ((k >> 2) * 6 + jt) * 64 + 2 * l + (k & 1)] = -0.5f * inv2;
                k = d + D;
                l = ((k & 3) >> 1) * 16 + li;
                sh[W_OFF + ((k >> 2) * 6 + jt) * 64 + 2 * l + (k & 1)] = mun * inv2;
            }
            sh[BIAS_OFF + j] = -0.5f * qc - logs - 58.81206412534085f;
        } else {
            for (int k = 0; k < 2 * D; ++k) {
                int l = ((k & 3) >> 1) * 16 + li;
                sh[W_OFF + ((k >> 2) * 6 + jt) * 64 + 2 * l + (k & 1)] = 0.f;
            }
            sh[BIAS_OFF + j] = 0.f;
        }
    }
    __syncthreads();

    // ---------- Phase 2: per-row LayerNorm(D) + L2 normalize, 2 threads per row
    {
        const int r = t & 127;           // row
        const int h = t >> 7;            // half: d in [32h, 32h+32)
        float* row = &sh[r * XT_STRIDE + 32 * h];
        float s1 = 0.f, s2 = 0.f;
        #pragma unroll 8
        for (int d = 0; d < 32; ++d) { float x = row[d]; s1 += x; s2 += x * x; }
        sh[EXA_OFF + t] = s1;
        sh[EXB_OFF + t] = s2;
        __syncthreads();
        s1 += sh[EXA_OFF + (t ^ 128)];
        s2 += sh[EXB_OFF + (t ^ 128)];
        float mean = s1 * (1.0f / D);
        float var  = s2 * (1.0f / D) - mean * mean;
        float rstd = rsqrtf(var + 1e-5f);
        float ssq = 0.f;
        #pragma unroll 8
        for (int d = 0; d < 32; ++d) {
            float y = (row[d] - mean) * rstd * sh[FW_OFF + 32 * h + d] + sh[FB_OFF + 32 * h + d];
            row[d] = y;
            ssq += y * y;
        }
        sh[EXC_OFF + t] = ssq;
        __syncthreads();
        ssq += sh[EXC_OFF + (t ^ 128)];
        float rn = 1.0f / fmaxf(sqrtf(ssq), 1e-12f);
        #pragma unroll 8
        for (int d = 0; d < 32; ++d) row[d] *= rn;
    }
    __syncthreads();

    // ---------- Phase 3: WMMA GEMM. Wave -> 16 pixels; 6 column tiles of 16 comps.
    // A frag (16x4 f32): lane l holds row m=l%16, k = 2*(l/16)+v. Frag reused
    // squared (k<64 half of W) and raw (k>=64 half). acc pre-seeded with bias.
    v8f acc[6];
    {
        const int c = lane & 15;
        #pragma unroll
        for (int j = 0; j < 6; ++j) {
            float bv = sh[BIAS_OFF + j * 16 + c];
            acc[j] = (v8f){bv, bv, bv, bv, bv, bv, bv, bv};
        }
    }
    {
        const int    nloc  = wv * 16 + (lane & 15);
        const int    khalf = (lane >> 4) * 2;
        const float* xrow  = &sh[nloc * XT_STRIDE];
        #pragma unroll 4
        for (int kk = 0; kk < 16; ++kk) {
            v2f a   = *(const v2f*)(xrow + kk * 4 + khalf);
            v2f asq = a * a;
            #pragma unroll
            for (int j = 0; j < 6; ++j) {
                const float* wp = &sh[W_OFF + (kk * 6 + j) * 64 + 2 * lane];
                v2f b1 = *(const v2f*)wp;                 // weights for x^2 (kstep kk)
                v2f b2 = *(const v2f*)(wp + 16 * 6 * 64); // weights for x   (kstep kk+16)
                acc[j] = __builtin_amdgcn_wmma_f32_16x16x4_f32(
                            false, asq, false, b1, (short)0, acc[j], false, false);
                acc[j] = __builtin_amdgcn_wmma_f32_16x16x4_f32(
                            false, a,   false, b2, (short)0, acc[j], false, false);
            }
        }
    }
    __syncthreads();   // everyone done reading xt/W -> reuse region for logp spill

    // ---------- Phase 4: spill logp tiles (C layout: pixel m = v+8*(l/16), comp = l%16)
    {
        float* lp = &sh[wv * (16 * LOGP_STRIDE)];
        const int ph = (lane >> 4) * 8;
        const int c  = lane & 15;
        #pragma unroll
        for (int j = 0; j < 6; ++j)
            #pragma unroll
            for (int v = 0; v < 8; ++v)
                lp[(ph + v) * LOGP_STRIDE + j * 16 + c] = acc[j][v];
    }
    __syncthreads();

    // ---------- Phase 5: max over P + LayerNorm(K) + transposed store.
    // Lane pair (p, p+16) shares pixel p: lanes<16 handle k=0..9, lanes>=16 k=10..18.
    {
        const int p  = lane & 15;
        const int h  = lane >> 4;
        const int k0 = 10 * h;
        const int nk = 10 - h;           // 10 or 9 classes
        const float* lp = &sh[wv * (16 * LOGP_STRIDE) + p * LOGP_STRIDE];
        float m[10];
        for (int i = 0; i < nk; ++i) {
            const float* g = lp + (k0 + i) * PCOMP;
            float mx = g[0];
            #pragma unroll
            for (int q = 1; q < PCOMP; ++q) mx = fmaxf(mx, g[q]);
            m[i] = mx;
        }
        float s1 = 0.f;
        for (int i = 0; i < nk; ++i) s1 += m[i];
        s1 += __shfl_xor(s1, 16, 32);                   // combine lane halves
        float mean = s1 * (1.0f / KCLS);
        float s2 = 0.f;
        for (int i = 0; i < nk; ++i) { float dd = m[i] - mean; s2 += dd * dd; }
        s2 += __shfl_xor(s2, 16, 32);
        float rstd = rsqrtf(s2 * (1.0f / KCLS) + 1e-5f);
        const int n = n0 + wv * 16 + p;
        float* op = out + (size_t)b * KCLS * Npix + n;
        for (int i = 0; i < nk; ++i) {
            int k = k0 + i;
            op[(size_t)k * Npix] = (m[i] - mean) * rstd * mask_w[k] + mask_b[k];
        }
    }
}

extern "C" void kernel_launch(void* const* d_in, const int* in_sizes, int n_in,
                              void* d_out, int out_size, void* d_ws, size_t ws_size,
                              hipStream_t stream) {
    const float* base   = (const float*)d_in[0];
    const float* means  = (const float*)d_in[1];
    const float* diagp  = (const float*)d_in[2];
    const float* feat_w = (const float*)d_in[3];
    const float* feat_b = (const float*)d_in[4];
    const float* mask_w = (const float*)d_in[5];
    const float* mask_b = (const float*)d_in[6];
    float* out = (float*)d_out;

    const int B = 8;
    const int Npix = in_sizes[0] / (B * D);         // 32768
    const int blocks = B * (Npix / TILE_N);         // 2048

    const int useWs = (d_ws != nullptr) &&
                      (ws_size >= (size_t)(W_FLOATS + NCOMP_P) * sizeof(float));
    if (useWs)
        build_w_kernel<<<1, 128, 0, stream>>>(means, diagp, (float*)d_ws);

    gmm_seg_head_kernel<<<blocks, THREADS, 0, stream>>>(
        base, means, diagp, feat_w, feat_b, mask_w, mask_b,
        (const float*)d_ws, out, Npix, useWs);
}